// EnsembleBeliefs_3642132267698
// MI455X (gfx1250) — compile-verified
//
#include <hip/hip_runtime.h>
#include <hip/hip_bf16.h>

// Problem constants (from reference): E=128 estimators, R=65536 regions, S=100000 samples.
#define E_EST 128
#define R_REG 65536
#define TPB   1024   // 32 wave32s per workgroup

// The async-to-LDS builtin takes pointers to 4x-int vectors (per hipcc's
// diagnostic in the previous round).
typedef __attribute__((ext_vector_type(4))) int v4i;

// One workgroup per (table t in {a,b}, estimator e). The full 256 KB estimator
// row is privatized in CDNA5's 320 KB LDS, so all scatter-adds become
// ds_add_f32 (LDS atomics) instead of contended L2 global atomics.
__global__ __launch_bounds__(TPB, 1)
void ensemble_beliefs_scatter(const float* __restrict__ a,
                              const float* __restrict__ b,
                              const int*   __restrict__ regions,  // [S, E] row-major
                              const float* __restrict__ da,       // [S]
                              const float* __restrict__ db,       // [S]
                              float*       __restrict__ out,      // [2, E, R]
                              int S)
{
    extern __shared__ float lds[];          // R_REG floats = 256 KB dynamic LDS

    const int t = blockIdx.x >> 7;          // 0 -> a-table, 1 -> b-table
    const int e = blockIdx.x & (E_EST - 1); // estimator id
    const int tid = threadIdx.x;

    const float* __restrict__ src  = (t == 0 ? a : b) + (size_t)e * R_REG;
    const float* __restrict__ dval = (t == 0 ? da : db);
    float*       __restrict__ dst  = out + ((size_t)(t * E_EST + e)) * R_REG;

    // ---- Stage the 256 KB estimator row global -> LDS ----------------------
#if defined(__gfx1250__) && __has_builtin(__builtin_amdgcn_global_load_async_to_lds_b128)
    // CDNA5 async DMA path: bypasses VGPRs, tracked by ASYNCcnt.
    for (int c = tid; c < R_REG / 4; c += TPB) {
        __builtin_amdgcn_global_load_async_to_lds_b128(
            (v4i*)(src + (size_t)c * 4),
            (v4i*)(lds + (size_t)c * 4),
            /*offset=*/0, /*cpol=*/0);
    }
  #if __has_builtin(__builtin_amdgcn_s_wait_asynccnt)
    __builtin_amdgcn_s_wait_asynccnt(0);
  #else
    asm volatile("s_wait_asynccnt 0" ::: "memory");
  #endif
#else
    // Portable fallback: vectorized b128 copy through VGPRs.
    for (int c = tid; c < R_REG / 4; c += TPB) {
        reinterpret_cast<float4*>(lds)[c] =
            reinterpret_cast<const float4*>(src)[c];
    }
#endif
    __syncthreads();

    // ---- Scatter-add all S samples into the LDS-resident row ---------------
    // Lane l of a wave handles sample s = base + l: da loads are coalesced;
    // index loads stride E*4 = 512B but 32 estimators share each 128B line and
    // the whole 51 MB index array lives in the 192 MB L2 across 256 blocks.
    for (int s = tid; s < S; s += TPB) {
        const int idx = regions[(size_t)s * E_EST + e];  // 0 <= idx < R
        const float v = dval[s];
        atomicAdd(&lds[idx], v);                         // -> ds_add_f32 (no return)
    }
    __syncthreads();

    // ---- Stream the updated row back out (also serves as out = a/b copy) ---
    for (int c = tid; c < R_REG / 4; c += TPB) {
        reinterpret_cast<float4*>(dst)[c] =
            reinterpret_cast<const float4*>(lds)[c];
    }
}

extern "C" void kernel_launch(void* const* d_in, const int* in_sizes, int n_in,
                              void* d_out, int out_size, void* d_ws, size_t ws_size,
                              hipStream_t stream) {
    const float* a       = (const float*)d_in[0];  // [E, R]
    const float* b       = (const float*)d_in[1];  // [E, R]
    const int*   regions = (const int*)  d_in[2];  // [S, E]
    const float* da      = (const float*)d_in[3];  // [S]
    const float* db      = (const float*)d_in[4];  // [S]
    float*       out     = (float*)d_out;          // [2, E, R]
    const int S = in_sizes[3];

    const size_t ldsBytes = (size_t)R_REG * sizeof(float);  // 262144 B < 320 KB WGP LDS

    // Opt in to >64KB dynamic LDS (host-side attribute set; idempotent,
    // not a stream op, safe under graph capture). Ignore failure.
    (void)hipFuncSetAttribute((const void*)ensemble_beliefs_scatter,
                              hipFuncAttributeMaxDynamicSharedMemorySize,
                              (int)ldsBytes);

    dim3 grid(2 * E_EST);   // 256 workgroups: (table, estimator)
    dim3 block(TPB);
    ensemble_beliefs_scatter<<<grid, block, ldsBytes, stream>>>(
        a, b, regions, da, db, out, S);
}